// Loss_20993800143146
// MI455X (gfx1250) — compile-verified
//
#include <hip/hip_runtime.h>
#include <math.h>

// ---- problem constants from the reference ----
#define H_CONST      4
#define N_CONST      3
#define DC_BW        50
#define STOPW        5.0f
#define DC_STRENGTH  1e-4f

typedef __attribute__((ext_vector_type(2))) float v2f;
typedef __attribute__((ext_vector_type(8))) float v8f;

static __device__ __forceinline__ int lane_id() { return (int)(threadIdx.x & 31u); }

// wave32 reductions via ds_swizzle (group-of-32 mode: offset = xor<<10 | or<<5 | and)
static __device__ __forceinline__ float wave_sum_f32(float v) {
  v += __int_as_float(__builtin_amdgcn_ds_swizzle(__float_as_int(v), 0x401F));
  v += __int_as_float(__builtin_amdgcn_ds_swizzle(__float_as_int(v), 0x201F));
  v += __int_as_float(__builtin_amdgcn_ds_swizzle(__float_as_int(v), 0x101F));
  v += __int_as_float(__builtin_amdgcn_ds_swizzle(__float_as_int(v), 0x081F));
  v += __int_as_float(__builtin_amdgcn_ds_swizzle(__float_as_int(v), 0x041F));
  return v;
}
static __device__ __forceinline__ int wave_sum_i32(int v) {
  v += __builtin_amdgcn_ds_swizzle(v, 0x401F);
  v += __builtin_amdgcn_ds_swizzle(v, 0x201F);
  v += __builtin_amdgcn_ds_swizzle(v, 0x101F);
  v += __builtin_amdgcn_ds_swizzle(v, 0x081F);
  v += __builtin_amdgcn_ds_swizzle(v, 0x041F);
  return v;
}
static __device__ __forceinline__ int wave_max_i32(int v) {
  v = max(v, __builtin_amdgcn_ds_swizzle(v, 0x401F));
  v = max(v, __builtin_amdgcn_ds_swizzle(v, 0x201F));
  v = max(v, __builtin_amdgcn_ds_swizzle(v, 0x101F));
  v = max(v, __builtin_amdgcn_ds_swizzle(v, 0x081F));
  v = max(v, __builtin_amdgcn_ds_swizzle(v, 0x041F));
  return v;
}

static __device__ __forceinline__ void block_partial_out(float acc, float* dst) {
  __shared__ float sw[8];
  float w = wave_sum_f32(acc);
  if (lane_id() == 0) sw[threadIdx.x >> 5] = w;
  __syncthreads();
  if (threadIdx.x == 0) {
    float s0 = 0.f;
    int nw = (int)(blockDim.x >> 5);
    for (int i = 0; i < nw; ++i) s0 += sw[i];
    *dst = s0;
  }
}

// One fused kernel; block role decided by blockIdx.x (block-uniform).
//  blocks [0, NA)          : banded alignment sum (only the nonzero band!)
//  blocks [NA, NA+NM)      : mel L1 sum
//  block  NA+NM            : stop BCE / mask.sum / lengths.sum / pad-zeroing
__global__ void k_main(const float* __restrict__ align,
                       const float4* __restrict__ p4,
                       const float4* __restrict__ g4,
                       const unsigned char* __restrict__ mask,
                       const float* __restrict__ stop_pred,
                       const int* __restrict__ lengths,
                       float* __restrict__ ws,
                       int NA, int NM, int NMpad,
                       int ROWS, int S, int T, int kk, int B,
                       long melChunks, int NMEL4, int btStep) {
  float* pa = ws;
  float* pm = ws + NA;
  float* sc = ws + NA + NMpad;
  int tid = (int)threadIdx.x;

  if ((int)blockIdx.x < NA) {
    // ---- banded-window sum: per (row, s) line the included t form one
    // contiguous interval ((s-BW)/k, (s+BW)/k]  (~2*BW/k elements) ----
    int wpb    = (int)(blockDim.x >> 5);
    int wgid   = (int)blockIdx.x * wpb + (tid >> 5);
    int nwaves = NA * wpb;
    int lane   = lane_id();
    int nLines = ROWS * S;
    float acc = 0.f;
    for (int line = wgid; line < nLines; line += nwaves) {
      int s   = line % S;
      int row = line / S;
      int b   = row % B;
      if (T < lengths[b]) continue;                 // bmask==0: wave-uniform skip
      int t_lo  = (s >= DC_BW) ? ((s - DC_BW) / kk + 1) : 0;
      int t_end = min((s + DC_BW) / kk + 1, T);
      const float* base = align + (long)line * T;
      for (int t = t_lo + lane; t < t_end; t += 32) acc += base[t];
    }
    block_partial_out(acc, pa + blockIdx.x);

  } else if ((int)blockIdx.x < NA + NM) {
    // ---- mel L1: grid chosen so stride % NMEL4 == 0 -> bt advances by a
    // constant; exactly one integer division per thread ----
    int mb     = (int)blockIdx.x - NA;
    int gid    = mb * (int)blockDim.x + tid;
    int stride = NM * (int)blockDim.x;
    int bt     = gid / NMEL4;
    float acc = 0.f;
    for (long c = gid; c < melChunks; c += stride, bt += btStep) {
      __builtin_prefetch(&p4[c + stride], 0, 1);    // global_prefetch_b8
      float m  = mask[bt] ? 1.f : 0.f;
      float4 p = p4[c];
      float4 g = g4[c];
      acc += fabsf(p.x * m - g.x) + fabsf(p.y * m - g.y) +
             fabsf(p.z * m - g.z) + fabsf(p.w * m - g.w);
    }
    block_partial_out(acc, pm + mb);

  } else {
    // ---- scalars + zero the WMMA pad of the mel partial array ----
    __shared__ float s_stop[64];
    __shared__ int   s_cnt;
    if (tid == 0) s_cnt = 0;
    for (int i = tid; i < 64; i += (int)blockDim.x) s_stop[i] = 0.f;
    for (int i = tid; i < NMpad - NM; i += (int)blockDim.x) pm[NM + i] = 0.f;
    __syncthreads();

    int cnt = 0;
    int n = B * T;
    for (int i = tid; i < n; i += (int)blockDim.x) cnt += mask[i] ? 1 : 0;
    cnt = wave_sum_i32(cnt);
    if (lane_id() == 0) atomicAdd(&s_cnt, cnt);

    int wid = tid >> 5, nw = (int)(blockDim.x >> 5);
    for (int b = wid; b < B; b += nw) {
      int mi = -1;
      for (int t = lane_id(); t < T; t += 32)
        if (mask[b * T + t]) mi = t;                // ascending t: last hit = lane max
      mi = wave_max_i32(mi);
      if (lane_id() == 0) {
        int idx  = mi < 0 ? 0 : mi;                 // jnp argmax of all-(-1) is 0
        float p  = stop_pred[b * T + idx];
        float lp = fmaxf(logf(p), -100.f);
        s_stop[b] = -STOPW * lp;
      }
    }
    __syncthreads();
    if (tid == 0) {
      float st = 0.f;
      for (int b = 0; b < B; ++b) st += s_stop[b];
      long ls = 0;
      for (int b = 0; b < B; ++b) ls += lengths[b];
      sc[0] = st;
      sc[1] = (float)s_cnt;
      sc[2] = (float)ls;
    }
  }
}

// ---- finalize: WMMA-reduce partials; D = A x ones + C => sum(D) = 16*sum(A) ----
__global__ void k_final(const float* __restrict__ ws, float* __restrict__ out,
                        int NA, int NMpad, float melDenom) {
  const float* pa = ws;
  const float* pm = ws + NA;
  const float* sc = ws + NA + NMpad;
  int lane = lane_id();

  v2f ones; ones.x = 1.f; ones.y = 1.f;
  v8f ca = {0.f,0.f,0.f,0.f,0.f,0.f,0.f,0.f};
  for (int g = 0; g < NA; g += 64) {
    v2f av; av.x = pa[g + 2 * lane]; av.y = pa[g + 2 * lane + 1];
    ca = __builtin_amdgcn_wmma_f32_16x16x4_f32(false, av, false, ones,
                                               (short)0, ca, false, false);
  }
  v8f cm = {0.f,0.f,0.f,0.f,0.f,0.f,0.f,0.f};
  for (int g = 0; g < NMpad; g += 64) {
    v2f av; av.x = pm[g + 2 * lane]; av.y = pm[g + 2 * lane + 1];
    cm = __builtin_amdgcn_wmma_f32_16x16x4_f32(false, av, false, ones,
                                               (short)0, cm, false, false);
  }
  float sa = 0.f, sm = 0.f;
  for (int i = 0; i < 8; ++i) { sa += ca[i]; sm += cm[i]; }
  sa = wave_sum_f32(sa) * (1.f / 16.f);
  sm = wave_sum_f32(sm) * (1.f / 16.f);

  if (threadIdx.x == 0) {
    float mel_loss  = sm / melDenom;
    float stop_loss = sc[0] / sc[1];
    float dc        = sa / ((float)(H_CONST * N_CONST) * sc[2]);
    out[0] = mel_loss + stop_loss - DC_STRENGTH * dc;
  }
}

extern "C" void kernel_launch(void* const* d_in, const int* in_sizes, int n_in,
                              void* d_out, int out_size, void* d_ws, size_t ws_size,
                              hipStream_t stream) {
  const int*           lengths     = (const int*)d_in[0];
  const unsigned char* mask        = (const unsigned char*)d_in[1];
  const float*         stop_pred   = (const float*)d_in[2];
  const float*         mels_pred   = (const float*)d_in[3];
  const float*         mels_target = (const float*)d_in[4];
  const float*         align       = (const float*)d_in[5];

  int  B    = in_sizes[0];
  long BT   = (long)in_sizes[2];
  int  T    = (int)(BT / B);
  long MELN = (long)in_sizes[3];
  int  NMEL = (int)(MELN / BT);
  long ALN  = (long)in_sizes[5];
  int  S    = (int)(ALN / ((long)N_CONST * B * H_CONST * T));
  int  kk   = T / S;
  int  ROWS = (int)(ALN / ((long)S * T));          // N * B * H

  const int NA    = 256;                            // align partials (4 WMMA groups)
  int       NMEL4 = NMEL / 4;                       // 20
  int       NM    = 12 * NMEL4;                     // 240 -> stride divisible by NMEL4
  int       NMpad = ((NM + 63) / 64) * 64;          // 256 (4 WMMA groups)
  int       btStep = (NM * 256) / NMEL4;            // bt advance per grid stride

  float* ws = (float*)d_ws;

  int nblocks = NA + NM + 1;
  k_main<<<nblocks, 256, 0, stream>>>(align, (const float4*)mels_pred,
                                      (const float4*)mels_target, mask, stop_pred,
                                      lengths, ws, NA, NM, NMpad,
                                      ROWS, S, T, kk, B,
                                      MELN / 4, NMEL4, btStep);
  k_final<<<1, 32, 0, stream>>>(ws, (float*)d_out, NA, NMpad, (float)MELN);
}